// GCN_62148176773352
// MI455X (gfx1250) — compile-verified
//
#include <hip/hip_runtime.h>

typedef float v2f __attribute__((ext_vector_type(2)));
typedef float v4f __attribute__((ext_vector_type(4)));
typedef float v8f __attribute__((ext_vector_type(8)));

// ---------------------------------------------------------------------------
// Dense GEMM:  out[M x NOUT] = (RELU_IN ? relu(A) : A)[M x K] @ W[K x NOUT] + bias
// One workgroup = 256 threads = 8 waves. Each wave computes one 16-row M-tile
// across all NOUT columns using V_WMMA_F32_16X16X4_F32.
//
// W is staged into LDS in k-pair-interleaved layout:
//   lds[(p*NOUT + n)*2 + j] = W[2p + j][n],   p = k/2
// so a B fragment (rows r, r+1 at column n) is one contiguous, 8B-aligned
// ds_load_b64 directly into an even-aligned VGPR pair (no register shuffles).
// ---------------------------------------------------------------------------
template <int K, int NOUT, bool RELU_IN>
__global__ __launch_bounds__(256) void gemm_wmma(const float* __restrict__ A,
                                                 const float* __restrict__ W,
                                                 const float* __restrict__ bias,
                                                 float* __restrict__ out,
                                                 int M) {
  extern __shared__ float ldsW[];  // K * NOUT floats, pair-interleaved
  const int tid = threadIdx.x;

  // Cooperative interleaved stage of W into LDS.
  // Each iteration handles (pair p, column pair n2): two float2 global reads,
  // one b128 LDS store of {W[2p][n], W[2p+1][n], W[2p][n+1], W[2p+1][n+1]}.
  {
    constexpr int NITER = (K / 2) * (NOUT / 2);
    for (int i = tid; i < NITER; i += 256) {
      const int p = i / (NOUT / 2);
      const int n = (i - p * (NOUT / 2)) * 2;
      const v2f r0 = *(const v2f*)(W + (2 * p + 0) * NOUT + n);
      const v2f r1 = *(const v2f*)(W + (2 * p + 1) * NOUT + n);
      v4f q;
      q.x = r0.x; q.y = r1.x; q.z = r0.y; q.w = r1.y;
      *(v4f*)(ldsW + (p * NOUT + n) * 2) = q;
    }
  }
  __syncthreads();

  const int wave = tid >> 5;
  const int lane = tid & 31;
  const int mtile = blockIdx.x * 8 + wave;
  if (mtile * 16 >= M) return;  // wave-uniform exit (EXEC stays all-ones)

  const int m0 = mtile * 16;
  const int arow_i = m0 + (lane & 15);  // A row for this lane
  const int khalf = (lane >> 4) << 1;   // 0 for lanes 0-15, 2 for 16-31
  const int ncol = lane & 15;           // N within a 16-col tile

  constexpr int NT = NOUT / 16;
  v8f acc[NT];
#pragma unroll
  for (int nt = 0; nt < NT; ++nt) {
    const float bv = bias[nt * 16 + ncol];
#pragma unroll
    for (int j = 0; j < 8; ++j) acc[nt][j] = bv;  // bias folded into C
  }

  const float* __restrict__ arow = A + (long long)arow_i * K;

  for (int k0 = 0; k0 < K; k0 += 4) {
    // A fragment: lane holds A[row][k0+khalf .. k0+khalf+1]  (global_load_b64)
    v2f a = *(const v2f*)(arow + k0 + khalf);
    if (RELU_IN) {
      a.x = fmaxf(a.x, 0.0f);
      a.y = fmaxf(a.y, 0.0f);
    }
    const int pbase = ((k0 + khalf) >> 1) * NOUT;  // pair index * NOUT
#pragma unroll
    for (int nt = 0; nt < NT; ++nt) {
      // B fragment: one ds_load_b64 of the interleaved pair at column n.
      const v2f b = *(const v2f*)(ldsW + (pbase + nt * 16 + ncol) * 2);
      acc[nt] = __builtin_amdgcn_wmma_f32_16x16x4_f32(
          /*neg_a=*/false, a, /*neg_b=*/false, b,
          /*c_mod=*/(short)0, acc[nt], /*reuse_a=*/false, /*reuse_b=*/false);
    }
  }

  // Store: C/D layout -> VGPR v, lanes 0-15: M=v, lanes 16-31: M=v+8; N=lane&15
  const int mrow = m0 + ((lane >> 4) << 3);
#pragma unroll
  for (int nt = 0; nt < NT; ++nt) {
#pragma unroll
    for (int v = 0; v < 8; ++v) {
      out[(long long)(mrow + v) * NOUT + nt * 16 + ncol] = acc[nt][v];
    }
  }
}

// ---------------------------------------------------------------------------
// Zero-init (float4 stores).
// ---------------------------------------------------------------------------
__global__ void zero_f4(float4* __restrict__ p, int n4) {
  int i = blockIdx.x * blockDim.x + threadIdx.x;
  if (i < n4) p[i] = make_float4(0.f, 0.f, 0.f, 0.f);
}

// ---------------------------------------------------------------------------
// COO SpMM scatter: out[dst[e], :] += val[e] * in[src[e], :]
// One thread per (edge, 4-float chunk): float4 gather + 4 fp32 atomic adds.
// ---------------------------------------------------------------------------
template <int F>
__global__ __launch_bounds__(256) void spmm_scatter(const int* __restrict__ src,
                                                    const int* __restrict__ dst,
                                                    const float* __restrict__ vals,
                                                    const float* __restrict__ in,
                                                    float* __restrict__ out,
                                                    int E) {
  constexpr int F4 = F / 4;
  const int tid = blockIdx.x * blockDim.x + threadIdx.x;
  const int e = tid / F4;
  if (e >= E) return;
  const int q = (tid - e * F4) * 4;

  const float val = vals[e];
  const float4 xin = *(const float4*)(in + (long long)src[e] * F + q);
  float* o = out + (long long)dst[e] * F + q;

  __hip_atomic_fetch_add(o + 0, val * xin.x, __ATOMIC_RELAXED, __HIP_MEMORY_SCOPE_AGENT);
  __hip_atomic_fetch_add(o + 1, val * xin.y, __ATOMIC_RELAXED, __HIP_MEMORY_SCOPE_AGENT);
  __hip_atomic_fetch_add(o + 2, val * xin.z, __ATOMIC_RELAXED, __HIP_MEMORY_SCOPE_AGENT);
  __hip_atomic_fetch_add(o + 3, val * xin.w, __ATOMIC_RELAXED, __HIP_MEMORY_SCOPE_AGENT);
}

// ---------------------------------------------------------------------------
// Row argmax over 16 logits (log_softmax is monotonic -> argmax of logits).
// First-max-wins to match jnp.argmax tie-breaking.
// ---------------------------------------------------------------------------
__global__ void argmax16(const float* __restrict__ L, int* __restrict__ out, int n) {
  const int i = blockIdx.x * blockDim.x + threadIdx.x;
  if (i >= n) return;
  const float* r = L + (long long)i * 16;
  int best = 0;
  float bv = r[0];
#pragma unroll
  for (int j = 1; j < 16; ++j) {
    const float v = r[j];
    if (v > bv) { bv = v; best = j; }
  }
  out[i] = best;
}

// ---------------------------------------------------------------------------
// Launch: gemm1 -> spmm1 -> gemm2(relu-in) -> spmm2 -> gemm3(relu-in) -> spmm3
//         -> argmax
// ---------------------------------------------------------------------------
extern "C" void kernel_launch(void* const* d_in, const int* in_sizes, int n_in,
                              void* d_out, int out_size, void* d_ws, size_t ws_size,
                              hipStream_t stream) {
  const float* x  = (const float*)d_in[0];
  const int*   ei = (const int*)d_in[1];
  const float* ev = (const float*)d_in[2];
  const float* W1 = (const float*)d_in[3];
  const float* b1 = (const float*)d_in[4];
  const float* W2 = (const float*)d_in[5];
  const float* b2 = (const float*)d_in[6];
  const float* W3 = (const float*)d_in[7];
  const float* b3 = (const float*)d_in[8];

  const int Mn = in_sizes[0] / 512;   // 100000 nodes
  const int E  = in_sizes[2];         // 1600000 edges
  const int* src = ei;                // edge_index[0]
  const int* dst = ei + E;            // edge_index[1]

  float* A = (float*)d_ws;                       // Mn*128 floats
  float* B = A + (size_t)Mn * 128;               // Mn*128 floats

  const int mtiles  = (Mn + 15) / 16;
  const int gblocks = (mtiles + 7) / 8;

  // Layer 1: T1 = X @ W1 + b1   (K=512, N=128; W1 staged: 256 KB LDS)
  gemm_wmma<512, 128, false><<<gblocks, 256, 512 * 128 * 4, stream>>>(x, W1, b1, A, Mn);
  {
    const int n4 = Mn * 128 / 4;
    zero_f4<<<(n4 + 255) / 256, 256, 0, stream>>>((float4*)B, n4);
  }
  {
    const long long t = (long long)E * (128 / 4);
    spmm_scatter<128><<<(int)((t + 255) / 256), 256, 0, stream>>>(src, dst, ev, A, B, E);
  }

  // Layer 2: T2 = relu(H1) @ W2 + b2   (K=128, N=64; 32 KB LDS)
  gemm_wmma<128, 64, true><<<gblocks, 256, 128 * 64 * 4, stream>>>(B, W2, b2, A, Mn);
  {
    const int n4 = Mn * 64 / 4;
    zero_f4<<<(n4 + 255) / 256, 256, 0, stream>>>((float4*)B, n4);
  }
  {
    const long long t = (long long)E * (64 / 4);
    spmm_scatter<64><<<(int)((t + 255) / 256), 256, 0, stream>>>(src, dst, ev, A, B, E);
  }

  // Layer 3: T3 = relu(H2) @ W3 + b3   (K=64, N=16; 4 KB LDS)
  gemm_wmma<64, 16, true><<<gblocks, 256, 64 * 16 * 4, stream>>>(B, W3, b3, A, Mn);
  {
    const int n4 = Mn * 16 / 4;
    zero_f4<<<(n4 + 255) / 256, 256, 0, stream>>>((float4*)B, n4);
  }
  {
    const long long t = (long long)E * (16 / 4);
    spmm_scatter<16><<<(int)((t + 255) / 256), 256, 0, stream>>>(src, dst, ev, A, B, E);
  }

  // Argmax of logits == argmax of log_softmax
  argmax16<<<(Mn + 255) / 256, 256, 0, stream>>>(B, (int*)d_out, Mn);
}